// GATModel_31361851195464
// MI455X (gfx1250) — compile-verified
//
#include <hip/hip_runtime.h>
#include <stdint.h>

// Problem constants (match reference setup_inputs)
#define NN 100000
#define EE 3200000LL
#define FIN 128
#define FE 16
#define CC 32
#define NEG_SLOPE 0.2f

typedef float v2f __attribute__((ext_vector_type(2)));
typedef float v8f __attribute__((ext_vector_type(8)));

// ---------- order-preserving float<->uint mapping for atomic max ----------
__device__ __forceinline__ unsigned ordf(float f) {
  unsigned u = __float_as_uint(f);
  return (u & 0x80000000u) ? ~u : (u | 0x80000000u);
}
__device__ __forceinline__ float unordf(unsigned u) {
  u = (u & 0x80000000u) ? (u & 0x7fffffffu) : ~u;
  return __uint_as_float(u);
}

// ---------- tiny precompute: w = We @ atte for both layers ----------
__global__ __launch_bounds__(32) void compute_w_k(
    const float* __restrict__ We1, const float* __restrict__ atte1,
    const float* __restrict__ We2, const float* __restrict__ atte2,
    float* __restrict__ w1, float* __restrict__ w2) {
  int t = threadIdx.x;
  if (t < FE) {
    float s = 0.f;
    for (int c = 0; c < CC; ++c) s += We1[t * CC + c] * atte1[c];
    w1[t] = s;
  } else if (t < 2 * FE) {
    int f = t - FE;
    float s = 0.f;
    for (int c = 0; c < CC; ++c) s += We2[f * CC + c] * atte2[c];
    w2[f] = s;
  }
}

__global__ void zero_k(float* __restrict__ p, long long n) {
  long long i = (long long)blockIdx.x * blockDim.x + threadIdx.x;
  if (i < n) p[i] = 0.f;
}

// ---------- segment sums of edge_attr over dst (for self-loop mean) ----------
__global__ void seg_sum_k(const long long* __restrict__ dstI,
                          const float* __restrict__ edge_attr,
                          float* __restrict__ s_sum, float* __restrict__ cnt) {
  long long e = (long long)blockIdx.x * blockDim.x + threadIdx.x;
  if (e >= EE) return;
  int d = (int)dstI[e];
  atomicAdd(&cnt[d], 1.f);
  const float* a = edge_attr + e * FE;
  float* sr = s_sum + (size_t)d * FE;
  for (int f = 0; f < FE; ++f) atomicAdd(&sr[f], a[f]);
}

// ---------- per-edge a_e for both layers in one pass over edge_attr ----------
__global__ void edge_ae_k(const float* __restrict__ edge_attr,
                          const float* __restrict__ w1, const float* __restrict__ w2,
                          float* __restrict__ ae1, float* __restrict__ ae2) {
  long long e = (long long)blockIdx.x * blockDim.x + threadIdx.x;
  if (e >= EE) return;
  const float* a = edge_attr + e * FE;
  float s1 = 0.f, s2 = 0.f;
  for (int f = 0; f < FE; ++f) {
    float v = a[f];
    s1 += v * w1[f];
    s2 += v * w2[f];
  }
  ae1[e] = s1;
  ae2[e] = s2;
}

// ---------- self-loop a_e (mean attr dotted with w) ----------
__global__ void loop_ae_k(const float* __restrict__ s_sum, const float* __restrict__ cnt,
                          const float* __restrict__ w1, const float* __restrict__ w2,
                          float* __restrict__ aeL1, float* __restrict__ aeL2) {
  int i = blockIdx.x * blockDim.x + threadIdx.x;
  if (i >= NN) return;
  float c = cnt[i];
  float s1 = 0.f, s2 = 0.f;
  if (c > 0.f) {
    float inv = 1.f / c;
    const float* sr = s_sum + (size_t)i * FE;
    for (int f = 0; f < FE; ++f) {
      float v = sr[f] * inv;
      s1 += v * w1[f];
      s2 += v * w2[f];
    }
  }
  aeL1[i] = s1;
  aeL2[i] = s2;
}

// ---------- WMMA GEMM: H[M x 32] = act(A[M x KDIM]) @ Wm[KDIM x 32] ----------
// One wave per 16-row tile; two 16x16 accumulators cover the 32 output cols.
// f32 WMMA (V_WMMA_F32_16X16X4_F32) keeps full precision vs. the f32 reference.
template <int KDIM, bool ACT>
__global__ __launch_bounds__(32) void gemm_wmma_k(
    const float* __restrict__ A, const float* __restrict__ Wm,
    const float* __restrict__ bias, float* __restrict__ H) {
  const int lane = threadIdx.x & 31;
  const int half = lane >> 4;   // 0 or 1
  const int l15 = lane & 15;
  const int m0 = blockIdx.x * 16;
  const int rowA = m0 + l15;
  const int kOff = half << 1;   // 0 or 2

  v8f acc0 = {};
  v8f acc1 = {};
  for (int k0 = 0; k0 < KDIM; k0 += 4) {
    float a0 = A[(size_t)rowA * KDIM + k0 + kOff];
    float a1 = A[(size_t)rowA * KDIM + k0 + kOff + 1];
    if (ACT) {  // fuse relu(x + b) from the previous layer into the A-fragment
      a0 = fmaxf(a0 + bias[k0 + kOff], 0.f);
      a1 = fmaxf(a1 + bias[k0 + kOff + 1], 0.f);
    }
    v2f av;
    av.x = a0;
    av.y = a1;
    v2f b0, b1;
    b0.x = Wm[(k0 + kOff) * CC + l15];
    b0.y = Wm[(k0 + kOff + 1) * CC + l15];
    b1.x = Wm[(k0 + kOff) * CC + 16 + l15];
    b1.y = Wm[(k0 + kOff + 1) * CC + 16 + l15];
    acc0 = __builtin_amdgcn_wmma_f32_16x16x4_f32(false, av, false, b0, (short)0,
                                                 acc0, false, false);
    acc1 = __builtin_amdgcn_wmma_f32_16x16x4_f32(false, av, false, b1, (short)0,
                                                 acc1, false, false);
  }
  // C/D layout: VGPR j -> row (j + 8*half), col = lane&15
  const int rBase = m0 + half * 8;
  for (int j = 0; j < 8; ++j) {
    H[(size_t)(rBase + j) * CC + l15] = acc0[j];
    H[(size_t)(rBase + j) * CC + 16 + l15] = acc1[j];
  }
}

// ---------- per-node: a_src/a_dst dots + init m/denom/out accumulators ----------
__global__ void node_prep_k(const float* __restrict__ H,
                            const float* __restrict__ att_s, const float* __restrict__ att_d,
                            float* __restrict__ a_src, float* __restrict__ a_dst,
                            unsigned* __restrict__ m_u, float* __restrict__ denom,
                            float* __restrict__ out_acc) {
  int i = blockIdx.x * blockDim.x + threadIdx.x;
  if (i >= NN) return;
  const float* hr = H + (size_t)i * CC;
  float s = 0.f, d = 0.f;
  for (int c = 0; c < CC; ++c) {
    float v = hr[c];
    s += v * att_s[c];
    d += v * att_d[c];
  }
  a_src[i] = s;
  a_dst[i] = d;
  m_u[i] = 0u;        // ordf-space minimum
  denom[i] = 0.f;
  float* o = out_acc + (size_t)i * CC;
  for (int c = 0; c < CC; ++c) o[c] = 0.f;
}

// ---------- alpha = leaky_relu(a_src[s]+a_dst[d]+a_e); segment max ----------
__global__ void alpha_max_k(const long long* __restrict__ srcI,
                            const long long* __restrict__ dstI,
                            const float* __restrict__ aeE, const float* __restrict__ aeL,
                            const float* __restrict__ a_src, const float* __restrict__ a_dst,
                            float* __restrict__ alpha, unsigned* __restrict__ m_u) {
  long long e = (long long)blockIdx.x * blockDim.x + threadIdx.x;
  const long long EN = EE + NN;
  if (e >= EN) return;
  int s, d;
  float ae;
  if (e < EE) {
    s = (int)srcI[e];
    d = (int)dstI[e];
    ae = aeE[e];
  } else {
    s = d = (int)(e - EE);
    ae = aeL[s];
  }
  float al = a_src[s] + a_dst[d] + ae;
  al = (al > 0.f) ? al : NEG_SLOPE * al;
  alpha[e] = al;
  atomicMax(&m_u[d], ordf(al));
}

// ---------- denom = segment sum of exp(alpha - m[dst]) ----------
__global__ void denom_k(const long long* __restrict__ dstI,
                        const float* __restrict__ alpha, const unsigned* __restrict__ m_u,
                        float* __restrict__ denom) {
  long long e = (long long)blockIdx.x * blockDim.x + threadIdx.x;
  const long long EN = EE + NN;
  if (e >= EN) return;
  int d = (e < EE) ? (int)dstI[e] : (int)(e - EE);
  float ea = __expf(alpha[e] - unordf(m_u[d]));
  atomicAdd(&denom[d], ea);
}

// ---------- out[dst] += h[src] * softmax coef ----------
__global__ void aggregate_k(const long long* __restrict__ srcI,
                            const long long* __restrict__ dstI,
                            const float* __restrict__ alpha, const unsigned* __restrict__ m_u,
                            const float* __restrict__ denom, const float* __restrict__ H,
                            float* __restrict__ out_acc) {
  long long e = (long long)blockIdx.x * blockDim.x + threadIdx.x;
  const long long EN = EE + NN;
  if (e >= EN) return;
  int s, d;
  if (e < EE) {
    s = (int)srcI[e];
    d = (int)dstI[e];
  } else {
    s = d = (int)(e - EE);
  }
  float coef = __expf(alpha[e] - unordf(m_u[d])) / denom[d];
  const float* hr = H + (size_t)s * CC;
  float* o = out_acc + (size_t)d * CC;
  for (int c = 0; c < CC; ++c) atomicAdd(&o[c], hr[c] * coef);
}

// ---------- final head: relu((out2 + b2) @ Wl + bl) ----------
__global__ void final_k(const float* __restrict__ out_acc, const float* __restrict__ b2,
                        const float* __restrict__ Wl, const float* __restrict__ bl,
                        float* __restrict__ out) {
  int i = blockIdx.x * blockDim.x + threadIdx.x;
  if (i >= NN) return;
  const float* o = out_acc + (size_t)i * CC;
  float acc = bl[0];
  for (int c = 0; c < CC; ++c) acc += (o[c] + b2[c]) * Wl[c];
  out[i] = fmaxf(acc, 0.f);
}

extern "C" void kernel_launch(void* const* d_in, const int* in_sizes, int n_in,
                              void* d_out, int out_size, void* d_ws, size_t ws_size,
                              hipStream_t stream) {
  const float* x = (const float*)d_in[0];
  const long long* ei = (const long long*)d_in[1];  // int64 [2, E]
  const float* edge_attr = (const float*)d_in[2];
  const float* W1 = (const float*)d_in[3];
  const float* att_src1 = (const float*)d_in[4];
  const float* att_dst1 = (const float*)d_in[5];
  const float* We1 = (const float*)d_in[6];
  const float* atte1 = (const float*)d_in[7];
  const float* b1 = (const float*)d_in[8];
  const float* W2 = (const float*)d_in[9];
  const float* att_src2 = (const float*)d_in[10];
  const float* att_dst2 = (const float*)d_in[11];
  const float* We2 = (const float*)d_in[12];
  const float* atte2 = (const float*)d_in[13];
  const float* b2 = (const float*)d_in[14];
  const float* Wl = (const float*)d_in[15];
  const float* bl = (const float*)d_in[16];
  float* out = (float*)d_out;

  const long long* srcI = ei;
  const long long* dstI = ei + EE;

  // ---- workspace carve-up (floats) ----
  float* ws = (float*)d_ws;
  size_t off = 0;
  float* w1 = ws + off;            off += 16;
  float* w2 = ws + off;            off += 48;           // pad to 64
  float* s_sum = ws + off;         off += (size_t)NN * FE;
  float* cnt = ws + off;           off += NN;
  float* ae1 = ws + off;           off += (size_t)EE;
  float* ae2 = ws + off;           off += (size_t)EE;
  float* aeL1 = ws + off;          off += NN;
  float* aeL2 = ws + off;          off += NN;
  float* h = ws + off;             off += (size_t)NN * CC;
  float* a_src = ws + off;         off += NN;
  float* a_dst = ws + off;         off += NN;
  float* alpha = ws + off;         off += (size_t)(EE + NN);
  unsigned* m_u = (unsigned*)(ws + off); off += NN;
  float* denom = ws + off;         off += NN;
  float* out_acc = ws + off;       off += (size_t)NN * CC;

  const int B = 256;
  const long long EN = EE + NN;
  dim3 gE((unsigned)((EE + B - 1) / B));
  dim3 gEN((unsigned)((EN + B - 1) / B));
  dim3 gN((NN + B - 1) / B);
  dim3 gTiles(NN / 16);  // 100000 / 16 = 6250 exactly

  // ---- precompute (shared by both layers) ----
  compute_w_k<<<1, 32, 0, stream>>>(We1, atte1, We2, atte2, w1, w2);
  {
    long long nz = (long long)NN * (FE + 1);  // s_sum and cnt are contiguous
    zero_k<<<(unsigned)((nz + B - 1) / B), B, 0, stream>>>(s_sum, nz);
  }
  seg_sum_k<<<gE, B, 0, stream>>>(dstI, edge_attr, s_sum, cnt);
  edge_ae_k<<<gE, B, 0, stream>>>(edge_attr, w1, w2, ae1, ae2);
  loop_ae_k<<<gN, B, 0, stream>>>(s_sum, cnt, w1, w2, aeL1, aeL2);

  // ---- layer 1 ----
  gemm_wmma_k<FIN, false><<<gTiles, 32, 0, stream>>>(x, W1, nullptr, h);
  node_prep_k<<<gN, B, 0, stream>>>(h, att_src1, att_dst1, a_src, a_dst, m_u, denom, out_acc);
  alpha_max_k<<<gEN, B, 0, stream>>>(srcI, dstI, ae1, aeL1, a_src, a_dst, alpha, m_u);
  denom_k<<<gEN, B, 0, stream>>>(dstI, alpha, m_u, denom);
  aggregate_k<<<gEN, B, 0, stream>>>(srcI, dstI, alpha, m_u, denom, h, out_acc);

  // ---- layer 2 (relu(out1 + b1) fused into GEMM A-fragment load) ----
  gemm_wmma_k<CC, true><<<gTiles, 32, 0, stream>>>(out_acc, W2, b1, h);
  node_prep_k<<<gN, B, 0, stream>>>(h, att_src2, att_dst2, a_src, a_dst, m_u, denom, out_acc);
  alpha_max_k<<<gEN, B, 0, stream>>>(srcI, dstI, ae2, aeL2, a_src, a_dst, alpha, m_u);
  denom_k<<<gEN, B, 0, stream>>>(dstI, alpha, m_u, denom);
  aggregate_k<<<gEN, B, 0, stream>>>(srcI, dstI, alpha, m_u, denom, h, out_acc);

  // ---- final linear + relu ----
  final_k<<<gN, B, 0, stream>>>(out_acc, b2, Wl, bl, out);
}